// SpatialGNNDynamicModel2_30399778521459
// MI455X (gfx1250) — compile-verified
//
#include <hip/hip_runtime.h>

#define N_NODES 100000
#define N_EDGES 1600000
#define FDIM    128
#define XS_PITCH 132   // 16B-aligned row pitch (528B) -> conflict-free ds_load_b64

typedef float v2f __attribute__((ext_vector_type(2)));
typedef float v8f __attribute__((ext_vector_type(8)));

// ---------------- degree / norm precompute ----------------

__global__ void deg_init_kernel(float* __restrict__ deg) {
    int i = blockIdx.x * blockDim.x + threadIdx.x;
    if (i < N_NODES) deg[i] = 1.0f;   // self-loop weight (fill_value=1.0)
}

__global__ void deg_accum_kernel(const int* __restrict__ col,
                                 const float* __restrict__ w,
                                 float* __restrict__ deg) {
    int e = blockIdx.x * blockDim.x + threadIdx.x;
    if (e < N_EDGES) atomicAdd(&deg[col[e]], w[e]);
}

__global__ void dinv_kernel(float* __restrict__ deg) {
    int i = blockIdx.x * blockDim.x + threadIdx.x;
    if (i < N_NODES) {
        float d = deg[i];
        deg[i] = (d > 0.0f) ? rsqrtf(d) : 0.0f;
    }
}

__global__ void norm_kernel(const int* __restrict__ row,
                            const int* __restrict__ col,
                            const float* __restrict__ w,
                            const float* __restrict__ dinv,
                            float* __restrict__ norm) {
    int e = blockIdx.x * blockDim.x + threadIdx.x;
    if (e < N_EDGES) norm[e] = dinv[row[e]] * w[e] * dinv[col[e]];
}

// ---------------- fp32 WMMA GEMM: H[N,128] = act(X)[N,128] @ W[128,128] ----
// Block = 256 threads = 8 waves; wave w owns output cols [16w, 16w+16).
// blockIdx.x owns output rows [16*bx, 16*bx+16). N_NODES % 16 == 0.
// The 16x128 A-tile is staged once through LDS (ReLU fused), shared by all
// 8 waves; A fragments come back via ds_load_b64 with a padded pitch so the
// 32-lane read covers all 64 LDS banks conflict-free.

__global__ __launch_bounds__(256) void gemm_wmma_kernel(
        const float* __restrict__ X, const float* __restrict__ W,
        float* __restrict__ H, int relu_in) {
    __shared__ float xs[16 * XS_PITCH];

    const int wave = threadIdx.x >> 5;
    const int lane = threadIdx.x & 31;
    const int half = lane >> 4;      // 0: lanes 0-15, 1: lanes 16-31
    const int l    = lane & 15;
    const int m0   = blockIdx.x * 16;
    const int n0   = wave * 16;

    // Cooperative stage: 16 rows x 32 float4 = 512 float4; 2 per thread.
    #pragma unroll
    for (int i = threadIdx.x; i < 16 * 32; i += 256) {
        const int r  = i >> 5;
        const int c4 = i & 31;
        float4 v = ((const float4*)(X + (size_t)(m0 + r) * FDIM))[c4];
        if (relu_in) {
            v.x = v.x > 0.0f ? v.x : 0.0f;
            v.y = v.y > 0.0f ? v.y : 0.0f;
            v.z = v.z > 0.0f ? v.z : 0.0f;
            v.w = v.w > 0.0f ? v.w : 0.0f;
        }
        *(float4*)(&xs[r * XS_PITCH + c4 * 4]) = v;   // 528B pitch keeps 16B align
    }
    __syncthreads();

    const float* arow = &xs[l * XS_PITCH];   // A row for this lane (LDS)
    const float* wcol = W + n0 + l;          // B col for this lane (global, L2-hot)

    v8f acc = {};
    #pragma unroll
    for (int k = 0; k < FDIM; k += 4) {
        const int kk = k + 2 * half;
        // A fragment: 16x4 f32, lane holds K=kk, kk+1 of row m0+l
        v2f a = *(const v2f*)(&arow[kk]);            // ds_load_b64
        // B fragment: 4x16 f32, lane holds rows K=kk, kk+1 at col n0+l
        v2f b = { wcol[(size_t)kk * FDIM], wcol[(size_t)(kk + 1) * FDIM] };
        acc = __builtin_amdgcn_wmma_f32_16x16x4_f32(
                /*neg_a=*/false, a, /*neg_b=*/false, b,
                /*c_mod=*/(short)0, acc, /*reuse_a=*/false, /*reuse_b=*/false);
    }

    // D: VGPR r, lane L -> row m0 + 8*half + r, col n0 + l
    float* hcol = H + n0 + l;
    const int mbase = m0 + 8 * half;
    #pragma unroll
    for (int r = 0; r < 8; ++r)
        hcol[(size_t)(mbase + r) * FDIM] = acc[r];
}

// ---------------- aggregation ----------------
// out[i] = bias + dinv[i]^2 * h[i]   (self-loop folded into init)

__global__ void agg_init_kernel(const float* __restrict__ H,
                                const float* __restrict__ bias,
                                const float* __restrict__ dinv,
                                float* __restrict__ out) {
    long idx = (long)blockIdx.x * blockDim.x + threadIdx.x;   // float4 index
    if (idx >= (long)N_NODES * (FDIM / 4)) return;
    int node = (int)(idx >> 5);        // FDIM/4 == 32 float4s per row
    int q    = (int)(idx & 31);
    float s = dinv[node]; s = s * s;
    float4 hv = ((const float4*)H)[idx];
    float4 bv = ((const float4*)bias)[q];
    float4 o;
    o.x = bv.x + s * hv.x;
    o.y = bv.y + s * hv.y;
    o.z = bv.z + s * hv.z;
    o.w = bv.w + s * hv.w;
    ((float4*)out)[idx] = o;
}

// out[col[e]] += norm[e] * h[row[e]]  — one wave per edge, float4 per lane.

__global__ __launch_bounds__(256) void scatter_kernel(
        const float* __restrict__ H,
        const int* __restrict__ row, const int* __restrict__ col,
        const float* __restrict__ norm, float* __restrict__ out) {
    long gw = ((long)blockIdx.x * blockDim.x + threadIdx.x) >> 5;
    int lane = threadIdx.x & 31;
    if (gw >= N_EDGES) return;
    int r = row[gw];
    int c = col[gw];
    float nv = norm[gw];
    float4 v = ((const float4*)(H + (size_t)r * FDIM))[lane];
    float* o = out + (size_t)c * FDIM + lane * 4;
    atomicAdd(o + 0, nv * v.x);
    atomicAdd(o + 1, nv * v.y);
    atomicAdd(o + 2, nv * v.z);
    atomicAdd(o + 3, nv * v.w);
}

// ---------------- launch ----------------

extern "C" void kernel_launch(void* const* d_in, const int* in_sizes, int n_in,
                              void* d_out, int out_size, void* d_ws, size_t ws_size,
                              hipStream_t stream) {
    const float* x  = (const float*)d_in[0];
    const int*   ei = (const int*)d_in[1];      // (2, E): row then col
    const float* ew = (const float*)d_in[2];
    const float* Ws[5] = {(const float*)d_in[3], (const float*)d_in[5],
                          (const float*)d_in[7], (const float*)d_in[9],
                          (const float*)d_in[11]};
    const float* bs[5] = {(const float*)d_in[4], (const float*)d_in[6],
                          (const float*)d_in[8], (const float*)d_in[10],
                          (const float*)d_in[12]};
    const int* row = ei;
    const int* col = ei + N_EDGES;

    float* out  = (float*)d_out;
    float* ws   = (float*)d_ws;
    float* dinv = ws;                               // N floats
    float* norm = ws + N_NODES;                     // E floats
    float* h    = ws + N_NODES + N_EDGES;           // N*128 floats (16B aligned)

    const int T = 256;
    deg_init_kernel <<<(N_NODES + T - 1) / T, T, 0, stream>>>(dinv);
    deg_accum_kernel<<<(N_EDGES + T - 1) / T, T, 0, stream>>>(col, ew, dinv);
    dinv_kernel     <<<(N_NODES + T - 1) / T, T, 0, stream>>>(dinv);
    norm_kernel     <<<(N_EDGES + T - 1) / T, T, 0, stream>>>(row, col, ew, dinv, norm);

    const long n_init4   = (long)N_NODES * (FDIM / 4);
    const long n_scatter = (long)N_EDGES * 32;

    for (int layer = 0; layer < 5; ++layer) {
        const float* in = (layer == 0) ? x : out;   // d_out ping-pongs as layer input
        gemm_wmma_kernel<<<N_NODES / 16, 256, 0, stream>>>(in, Ws[layer], h,
                                                           layer > 0 ? 1 : 0);
        agg_init_kernel <<<(int)((n_init4 + T - 1) / T), T, 0, stream>>>(h, bs[layer],
                                                                         dinv, out);
        scatter_kernel  <<<(int)((n_scatter + T - 1) / T), T, 0, stream>>>(h, row, col,
                                                                           norm, out);
    }
    // Final layer has no ReLU: d_out already holds the result.
}